// pna_73177652789899
// MI455X (gfx1250) — compile-verified
//
#include <hip/hip_runtime.h>
#include <math.h>

// PNA aggregation + 2-layer MLP for MI455X (gfx1250, wave32, WMMA).
// B=256, N=1024, D=512, H=2048, O=512. fp32 throughout (reference precision).
// GEMMs: V_WMMA_F32_16X16X4_F32 with double-buffered ASYNC global->LDS staging.

#define B_  256
#define N_  1024
#define D_  512
#define H_  2048
#define O_  512

typedef __attribute__((ext_vector_type(2))) float v2f;
typedef __attribute__((ext_vector_type(8))) float v8f;

// ---------------------------------------------------------------------------
// CDNA5 async global->LDS copy (ASYNCcnt) helpers
// ---------------------------------------------------------------------------
__device__ __forceinline__ void async_load_b128(float* lds_dst, const float* gsrc) {
    // low 32 bits of a generic pointer to LDS == wave-relative LDS byte offset
    unsigned lds_off = (unsigned)(size_t)lds_dst;
    asm volatile("global_load_async_to_lds_b128 %0, %1, off"
                 :: "v"(lds_off), "v"(gsrc)
                 : "memory");
}
__device__ __forceinline__ void wait_async0() {
    asm volatile("s_wait_asynccnt 0x0" ::: "memory");
}

// ---------------------------------------------------------------------------
// Kernel 1: masked per-batch aggregation (mean/max/min/std) over N.
// One workgroup (512 threads) per batch; mask row staged in LDS; one pass.
// Writes rows [0,4B) of the scale_concat buffer.
// ---------------------------------------------------------------------------
__global__ __launch_bounds__(512) void agg_kernel(
    const float* __restrict__ x,     // [B,N,D]
    const float* __restrict__ mask,  // [B,N,1]
    float* __restrict__ scale)       // [12B,D], rows 0..4B-1 written here
{
    __shared__ float mrow[N_];
    const int b = blockIdx.x;
    const int d = threadIdx.x;                 // 512 threads == D_

    for (int n = threadIdx.x; n < N_; n += 512) mrow[n] = mask[(size_t)b * N_ + n];
    __syncthreads();

    float sum = 0.f, sumsq = 0.f, cnt = 0.f;
    float mx = -INFINITY, mn = INFINITY;
    const float* xp = x + (size_t)b * N_ * D_ + d;
    #pragma unroll 4
    for (int n = 0; n < N_; ++n) {
        const float m  = mrow[n];
        const float xv = xp[(size_t)n * D_];
        sum   += m * xv;
        sumsq += m * xv * xv;
        cnt   += m;
        mx = fmaxf(mx, (m > 0.f) ? xv : -INFINITY);
        mn = fminf(mn, (m > 0.f) ? xv :  INFINITY);
    }
    const float mean = sum / cnt;
    const float var  = fmaxf(sumsq / cnt - mean * mean, 0.f);
    scale[(size_t)(0 * B_ + b) * D_ + d] = mean;
    scale[(size_t)(1 * B_ + b) * D_ + d] = mx;
    scale[(size_t)(2 * B_ + b) * D_ + d] = mn;
    scale[(size_t)(3 * B_ + b) * D_ + d] = sqrtf(var);
}

// ---------------------------------------------------------------------------
// Kernel 2: delta = sum(unpadded) + 1  (unpadded.shape[1] == 1)
// ---------------------------------------------------------------------------
__global__ __launch_bounds__(256) void delta_kernel(
    const float* __restrict__ unpadded,  // [4B]
    float* __restrict__ delta)
{
    __shared__ float red[256];
    const int tid = threadIdx.x;
    float s = 0.f;
    for (int i = tid; i < 4 * B_; i += 256) s += unpadded[i];
    red[tid] = s;
    __syncthreads();
    for (int off = 128; off > 0; off >>= 1) {
        if (tid < off) red[tid] += red[tid + off];
        __syncthreads();
    }
    if (tid == 0) *delta = red[0] + 1.0f;
}

// ---------------------------------------------------------------------------
// Kernel 3: rows [4B,8B) = aggr*(logu/delta); rows [8B,12B) = aggr*(delta/logu)
// ---------------------------------------------------------------------------
__global__ __launch_bounds__(256) void scale_kernel(
    const float* __restrict__ unpadded,
    const float* __restrict__ delta_p,
    float* __restrict__ scale)           // [12B,D]
{
    const int idx = blockIdx.x * 256 + threadIdx.x;   // over 4B*D
    const int r   = idx >> 9;                         // / D_ (512)
    const int d   = idx & (D_ - 1);
    const float delta = *delta_p;
    const float a     = scale[(size_t)r * D_ + d];
    const float lu    = logf(unpadded[r] + 1.0f);
    scale[(size_t)(4 * B_ + r) * D_ + d] = a * (lu / delta);
    scale[(size_t)(8 * B_ + r) * D_ + d] = a * (delta / lu);
}

// ---------------------------------------------------------------------------
// Kernel 4: 32x32 tiled transpose  dst[N][K] = src[K][N]
// ---------------------------------------------------------------------------
__global__ __launch_bounds__(256) void transpose_kernel(
    const float* __restrict__ src, float* __restrict__ dst, int K, int N)
{
    __shared__ float t[32][33];
    const int bk = blockIdx.y * 32;
    const int bn = blockIdx.x * 32;
    const int tx = threadIdx.x & 31;
    const int ty = threadIdx.x >> 5;      // 8 rows per pass
    #pragma unroll
    for (int i = 0; i < 32; i += 8)
        t[ty + i][tx] = src[(size_t)(bk + ty + i) * N + bn + tx];
    __syncthreads();
    #pragma unroll
    for (int i = 0; i < 32; i += 8)
        dst[(size_t)(bn + ty + i) * K + bk + tx] = t[tx][ty + i];
}

// ---------------------------------------------------------------------------
// GEMM: C[M,N] = act(A[M,K] @ B[K,N] + bias[N]), B supplied TRANSPOSED
// (Bt[N][K]) so both LDS tiles are [rows][K-slice] and stage via contiguous
// GLOBAL_LOAD_ASYNC_TO_LDS_B128, double-buffered.
// Block tile 64(M) x 128(N), KB=32; 8 waves = 4(M) x 2(N); wave = 16x64.
// Stride 36 floats: 16B-aligned b128 LDS writes, conflict-free b64 frags.
// ---------------------------------------------------------------------------
#define KB   32
#define LDST 36   // padded tile stride (floats)

template <int ROWS>
__device__ __forceinline__ void stage_async(const float* __restrict__ g, int ld,
                                            float* s, int tid) {
    // tile: ROWS x KB floats; g points at (row0, kb); s stride LDST
    #pragma unroll
    for (int i = 0; i < (ROWS * 8) / 256; ++i) {
        const int idx4 = tid + i * 256;
        const int row  = idx4 >> 3;           // 8 float4 per 32-wide row
        const int c    = (idx4 & 7) * 4;
        async_load_b128(s + row * LDST + c, g + (size_t)row * ld + c);
    }
}

template <bool RELU>
__global__ __launch_bounds__(256) void gemm_wmma_kernel(
    const float* __restrict__ A,    // [M,K]
    const float* __restrict__ Bt,   // [N,K]  (pre-transposed W)
    const float* __restrict__ bias, // [N]
    float* __restrict__ C,          // [M,N]
    int M, int N, int K)
{
    __shared__ float As[2][64][LDST];
    __shared__ float Bs[2][128][LDST];

    const int tid  = threadIdx.x;
    const int lane = tid & 31;
    const int wave = tid >> 5;
    const int wm   = wave & 3;    // 0..3 -> M sub-tile
    const int wn   = wave >> 2;   // 0..1 -> N half
    const int blockM = blockIdx.y * 64;
    const int blockN = blockIdx.x * 128;

    const int ml    = lane & 15;          // row/col within 16x16 tile
    const int khalf = (lane >> 4) * 2;    // ISA 16x4 f32 A/B K-pair selector

    const float* Ag = A  + (size_t)blockM * K;
    const float* Bg = Bt + (size_t)blockN * K;

    v8f acc[4] = {};

    // prologue: stage kb=0 into buffer 0
    stage_async<64>(Ag, K, &As[0][0][0], tid);
    stage_async<128>(Bg, K, &Bs[0][0][0], tid);

    int buf = 0;
    for (int kb = 0; kb < K; kb += KB, buf ^= 1) {
        wait_async0();        // this wave's pending copies into `buf` done
        __syncthreads();      // everyone's copies visible; prev reads done

        if (kb + KB < K) {    // prefetch next K-block into other buffer
            stage_async<64>(Ag + kb + KB, K, &As[buf ^ 1][0][0], tid);
            stage_async<128>(Bg + kb + KB, K, &Bs[buf ^ 1][0][0], tid);
        }

        const float* Ab = &As[buf][wm * 16 + ml][khalf];
        const float* Bb = &Bs[buf][wn * 64 + ml][khalf];
        #pragma unroll
        for (int k0 = 0; k0 < KB; k0 += 4) {
            const v2f a = *(const v2f*)(Ab + k0);          // ds_load_b64
            #pragma unroll
            for (int nt = 0; nt < 4; ++nt) {
                const v2f b = *(const v2f*)(Bb + nt * 16 * LDST + k0);
                acc[nt] = __builtin_amdgcn_wmma_f32_16x16x4_f32(
                    false, a, false, b, (short)0, acc[nt], false, false);
            }
        }
    }

    // epilogue: bias (+ReLU), store. C layout: VGPR r -> M=r (lanes 0-15),
    // M=r+8 (lanes 16-31); N = lane&15.
    const int rowBase = blockM + wm * 16 + ((lane >> 4) << 3);
    const int colBase = blockN + wn * 64 + ml;
    #pragma unroll
    for (int nt = 0; nt < 4; ++nt) {
        const int col = colBase + nt * 16;
        const float bv = bias[col];
        #pragma unroll
        for (int r = 0; r < 8; ++r) {
            float v = acc[nt][r] + bv;
            if (RELU) v = fmaxf(v, 0.f);
            C[(size_t)(rowBase + r) * N + col] = v;
        }
    }
}

// ---------------------------------------------------------------------------
// Launch
// ---------------------------------------------------------------------------
extern "C" void kernel_launch(void* const* d_in, const int* in_sizes, int n_in,
                              void* d_out, int out_size, void* d_ws, size_t ws_size,
                              hipStream_t stream) {
    const float* x        = (const float*)d_in[0];
    const float* mask     = (const float*)d_in[1];
    const float* unpadded = (const float*)d_in[2];
    const float* W1       = (const float*)d_in[3];
    const float* b1       = (const float*)d_in[4];
    const float* W2       = (const float*)d_in[5];
    const float* b2       = (const float*)d_in[6];
    float* out = (float*)d_out;

    // ws layout (floats): scale[12B*D] | h[12B*H] | W1t[H*D] | W2t[O*H] | delta
    float* scale = (float*)d_ws;
    float* h     = scale + (size_t)12 * B_ * D_;
    float* W1t   = h     + (size_t)12 * B_ * H_;
    float* W2t   = W1t   + (size_t)H_ * D_;
    float* delta = W2t   + (size_t)O_ * H_;

    // 1) masked aggregation -> rows [0,4B) of scale_concat
    agg_kernel<<<B_, 512, 0, stream>>>(x, mask, scale);

    // 2) delta scalar
    delta_kernel<<<1, 256, 0, stream>>>(unpadded, delta);

    // 3) amplification / attenuation rows
    scale_kernel<<<(4 * B_ * D_) / 256, 256, 0, stream>>>(unpadded, delta, scale);

    // 4) pre-transpose weights (keeps GEMM async staging contiguous)
    {
        dim3 g1(H_ / 32, D_ / 32);
        transpose_kernel<<<g1, 256, 0, stream>>>(W1, W1t, D_, H_);
        dim3 g2(O_ / 32, H_ / 32);
        transpose_kernel<<<g2, 256, 0, stream>>>(W2, W2t, H_, O_);
    }

    // 5) h = relu(scale_concat @ W1 + b1)   [3072,512]x[512,2048]
    {
        dim3 grid(H_ / 128, (12 * B_) / 64);
        gemm_wmma_kernel<true><<<grid, 256, 0, stream>>>(scale, W1t, b1, h,
                                                         12 * B_, H_, D_);
    }
    // 6) out = h @ W2 + b2                  [3072,2048]x[2048,512]
    {
        dim3 grid(O_ / 128, (12 * B_) / 64);
        gemm_wmma_kernel<false><<<grid, 256, 0, stream>>>(h, W2t, b2, out,
                                                          12 * B_, O_, H_);
    }
}